// Full_Hypernetwork_6786048328001
// MI455X (gfx1250) — compile-verified
//
#include <hip/hip_runtime.h>
#include <hip/hip_bf16.h>

// ---------------------------------------------------------------------------
// Full hypernetwork, fused:  p = h2 @ Wh3 never hits HBM.
//   B=512, IN=64, H=256, L=3, OUT=128, HYP=512, TOTAL=230272
// Dominant cost: one streaming pass over Wh3 (471 MB) ~= 20us @ 23.3 TB/s.
// ---------------------------------------------------------------------------

typedef __attribute__((ext_vector_type(2))) float v2f;
typedef __attribute__((ext_vector_type(8))) float v8f;
typedef __attribute__((ext_vector_type(4))) int   v4i;

#define TOTAL_COLS 230272L
#define A_STR 516   // LDS pitch for 64x512 A tile (conflict-free b64 A-frag reads)
#define B_STR 40    // LDS pitch for 128x32 B panel
#define H_STR 36    // LDS pitch for 64x32 h tile (keeps float4 alignment)

// ---- async global -> LDS path (CDNA5), with fallback ----------------------
#if defined(__has_builtin)
#if __has_builtin(__builtin_amdgcn_global_load_async_to_lds_b128)
#define USE_ASYNC_LDS 1
#endif
#endif
#ifndef USE_ASYNC_LDS
#define USE_ASYNC_LDS 0
#endif

#if defined(__has_builtin)
#if __has_builtin(__builtin_amdgcn_s_wait_asynccnt)
#define WAIT_ASYNC(n) __builtin_amdgcn_s_wait_asynccnt(n)
#endif
#endif
#ifndef WAIT_ASYNC
#define WAIT_ASYNC(n) asm volatile("s_wait_asynccnt %0" ::"n"(n) : "memory")
#endif

#if USE_ASYNC_LDS
// Issue the 4 per-thread 16B async copies of one 128x32 panel.
__device__ __forceinline__ void asyncB(const float4* __restrict__ W4, long base4,
                                       long rowStride4, float* __restrict__ Bp,
                                       int t)
{
  const int kloc0 = t >> 3;
  const int jq    = t & 7;
#pragma unroll
  for (int q = 0; q < 4; ++q) {
    const float4* g = W4 + base4 + (long)(kloc0 + 32 * q) * rowStride4 + jq;
    float*        l = Bp + (kloc0 + 32 * q) * B_STR + jq * 4;
    __builtin_amdgcn_global_load_async_to_lds_b128(
        (v4i*)const_cast<float4*>(g), (v4i*)l, 0, 0);
  }
}
#endif

// ---- one 128-deep K panel of WMMA fp32 16x16x4 ----------------------------
__device__ __forceinline__ v8f kpanel128(const float* __restrict__ Ap,
                                         const float* __restrict__ Bp,
                                         int rowA, int hi, int nC, v8f P)
{
#pragma unroll
  for (int ks = 0; ks < 32; ++ks) {
    const int k = ks * 4 + 2 * hi;
    v2f a; a.x = Ap[rowA * A_STR + k];  a.y = Ap[rowA * A_STR + k + 1];
    v2f b; b.x = Bp[k * B_STR + nC];    b.y = Bp[(k + 1) * B_STR + nC];
    P = __builtin_amdgcn_wmma_f32_16x16x4_f32(false, a, false, b, (short)0, P,
                                              false, false);
  }
  return P;
}

// ---- B-panel prefetch fallback (regs + ds_store) --------------------------
struct PanelRegs { float4 v[4]; };

__device__ __forceinline__ PanelRegs fetchB(const float4* __restrict__ W4,
                                            long base4, long rowStride4, int t)
{
  PanelRegs r;
  const int kloc0 = t >> 3;
  const int jq    = t & 7;
#pragma unroll
  for (int q = 0; q < 4; ++q)
    r.v[q] = W4[base4 + (long)(kloc0 + 32 * q) * rowStride4 + jq];
  return r;
}

__device__ __forceinline__ void storeB(float* __restrict__ Bp, int t,
                                       const PanelRegs& r)
{
  const int kloc0 = t >> 3;
  const int jq    = t & 7;
#pragma unroll
  for (int q = 0; q < 4; ++q)
    *reinterpret_cast<float4*>(Bp + (kloc0 + 32 * q) * B_STR + jq * 4) = r.v[q];
}

// ---------------------------------------------------------------------------
// feat = relu(x[:, :63] @ Wf + bf)                     (tiny, VALU)
// ---------------------------------------------------------------------------
__global__ __launch_bounds__(256) void k_feat(const float* __restrict__ x,
                                              const float* __restrict__ Wf,
                                              const float* __restrict__ bf,
                                              float* __restrict__ feat)
{
  __shared__ float xr[64];
  const int b = blockIdx.x, j = threadIdx.x;
  if (j < 64) xr[j] = x[b * 64 + j];
  __syncthreads();
  float acc = bf[j];
  for (int k = 0; k < 63; ++k)
    acc = fmaf(xr[k], Wf[k * 256 + j], acc);
  feat[b * 256 + j] = fmaxf(acc, 0.f);
}

// ---------------------------------------------------------------------------
// h2 = relu(h1 @ Wh2 + bh2), with h1 = relu(tau * Wh1 + bh1) built on the fly
// ---------------------------------------------------------------------------
__global__ __launch_bounds__(256) void k_h2(const float* __restrict__ x,
                                            const float* __restrict__ Wh1,
                                            const float* __restrict__ bh1,
                                            const float* __restrict__ Wh2,
                                            const float* __restrict__ bh2,
                                            float* __restrict__ h2)
{
  extern __shared__ float sm[];
  float* As   = sm;                     // 64 * A_STR
  float* Bs   = As + 64 * A_STR;        // 2 * 128 * B_STR
  float* taus = Bs + 2 * 128 * B_STR;   // 64

  const int t  = threadIdx.x;
  const int b0 = blockIdx.x * 64;
  const int j0 = blockIdx.y * 32;

  if (t < 64) taus[t] = x[(b0 + t) * 64 + 63];
  __syncthreads();

  {
    const float4* w1 = reinterpret_cast<const float4*>(Wh1);
    const float4* bb = reinterpret_cast<const float4*>(bh1);
    for (int e = t; e < 64 * 128; e += 256) {
      const int r = e >> 7, kq = e & 127;
      const float tau = taus[r];
      const float4 w = w1[kq], b = bb[kq];
      float4 h;
      h.x = fmaxf(fmaf(tau, w.x, b.x), 0.f);
      h.y = fmaxf(fmaf(tau, w.y, b.y), 0.f);
      h.z = fmaxf(fmaf(tau, w.z, b.z), 0.f);
      h.w = fmaxf(fmaf(tau, w.w, b.w), 0.f);
      *reinterpret_cast<float4*>(As + r * A_STR + kq * 4) = h;
    }
  }

  const int lane = t & 31, wv = t >> 5;
  const int rs = wv & 3, cs = wv >> 2;
  const int rowA = rs * 16 + (lane & 15);
  const int hi   = lane >> 4;
  const int nC   = cs * 16 + (lane & 15);

  const float4* W4 = reinterpret_cast<const float4*>(Wh2);
  const long rowStride4 = 512 / 4;
  const long base4      = j0 >> 2;

  v8f P = {0.f, 0.f, 0.f, 0.f, 0.f, 0.f, 0.f, 0.f};

#if USE_ASYNC_LDS
  asyncB(W4, base4, rowStride4, Bs, t);
  for (int kp = 0; kp < 4; ++kp) {
    float* Bp = Bs + (kp & 1) * 128 * B_STR;
    if (kp < 3) {
      asyncB(W4, base4 + (long)(kp + 1) * 128 * rowStride4, rowStride4,
             Bs + ((kp + 1) & 1) * 128 * B_STR, t);
      WAIT_ASYNC(4);
    } else {
      WAIT_ASYNC(0);
    }
    __syncthreads();
    P = kpanel128(As + kp * 128, Bp, rowA, hi, nC, P);
    __syncthreads();
  }
#else
  PanelRegs pre = fetchB(W4, base4, rowStride4, t);
  for (int kp = 0; kp < 4; ++kp) {
    float* Bp = Bs + (kp & 1) * 128 * B_STR;
    __syncthreads();
    storeB(Bp, t, pre);
    __syncthreads();
    if (kp < 3)
      pre = fetchB(W4, base4 + (long)(kp + 1) * 128 * rowStride4, rowStride4, t);
    P = kpanel128(As + kp * 128, Bp, rowA, hi, nC, P);
  }
#endif

  const float bj = bh2[j0 + nC];
#pragma unroll
  for (int e = 0; e < 8; ++e) {
    const int row = b0 + rs * 16 + e + 8 * hi;
    h2[(long)row * 512 + (j0 + nC)] = fmaxf(P[e] + bj, 0.f);
  }
}

// ---------------------------------------------------------------------------
// bvec[b,j] = h2[b,:] . Wh3[:, offb+j] + bh3[offb+j]   (per-layer bias column)
// ---------------------------------------------------------------------------
__global__ __launch_bounds__(256) void k_bvec(const float* __restrict__ h2,
                                              const float* __restrict__ Wh3,
                                              const float* __restrict__ bh3,
                                              long offb, int Hout,
                                              float* __restrict__ bvec)
{
  __shared__ float h2r[512];
  const int b = blockIdx.x, t = threadIdx.x;
  for (int k = t; k < 512; k += 256) h2r[k] = h2[(long)b * 512 + k];
  __syncthreads();
  for (int j = t; j < Hout; j += 256) {
    float acc = bh3[offb + j];
    for (int k = 0; k < 512; ++k)
      acc = fmaf(h2r[k], Wh3[(long)k * TOTAL_COLS + offb + j], acc);
    bvec[(long)b * Hout + j] = acc;
  }
}

// ---------------------------------------------------------------------------
// Fused layer: for each i in a 32-wide chunk:
//   P[b,j] = h2[b,:] @ Wh3[:, offW + i*Hout + j]      (WMMA fp32 16x16x4)
//   acc[b,j] += h_in[b,i] * (P[b,j] + bh3[offW + i*Hout + j])
// Writes deterministic partial sums: part[chunk][b][j].
// ---------------------------------------------------------------------------
__global__ __launch_bounds__(256) void k_layer(const float* __restrict__ h2,
                                               const float* __restrict__ hin,
                                               const float* __restrict__ Wh3,
                                               const float* __restrict__ bh3,
                                               float* __restrict__ part,
                                               int Hin, int Hout, long offW)
{
  extern __shared__ float sm[];
  float* As  = sm;                     // 64 * A_STR      (h2 tile, all K)
  float* Bs  = As + 64 * A_STR;        // 2 * 128 * B_STR (Wh3 double buffer)
  float* Hs  = Bs + 2 * 128 * B_STR;   // 64 * H_STR      (h_in tile)
  float* bts = Hs + 64 * H_STR;        // 32 * 32         (bh3 tile)

  const int t  = threadIdx.x;
  const int b0 = blockIdx.x * 64;
  const int j0 = blockIdx.y * 32;
  const int i0 = blockIdx.z * 32;

  {  // stage A = h2 tile (64 x 512)
    const float4* h2v = reinterpret_cast<const float4*>(h2);
    for (int e = t; e < 64 * 128; e += 256) {
      const int r = e >> 7, kq = e & 127;
      const float4 v = h2v[(long)(b0 + r) * 128 + kq];
      *reinterpret_cast<float4*>(As + r * A_STR + kq * 4) = v;
    }
  }
  {  // stage h_in tile (64 x 32)
    const float4* hv = reinterpret_cast<const float4*>(hin);
    for (int e = t; e < 64 * 8; e += 256) {
      const int r = e >> 3, iq = e & 7;
      const float4 v = hv[((long)(b0 + r) * Hin + i0) / 4 + iq];
      *reinterpret_cast<float4*>(Hs + r * H_STR + iq * 4) = v;
    }
  }
  {  // stage bh3 tile (32 x 32)
    const float4* bv = reinterpret_cast<const float4*>(bh3);
    for (int e = t; e < 32 * 8; e += 256) {
      const int ii = e >> 3, jq = e & 7;
      const float4 v = bv[(offW + (long)(i0 + ii) * Hout + j0) / 4 + jq];
      *reinterpret_cast<float4*>(bts + ii * 32 + jq * 4) = v;
    }
  }
  __syncthreads();

  const int lane = t & 31, wv = t >> 5;
  const int rs = wv & 3, cs = wv >> 2;
  const int rowA = rs * 16 + (lane & 15);
  const int hi   = lane >> 4;
  const int nC   = cs * 16 + (lane & 15);

  const float4* W4 = reinterpret_cast<const float4*>(Wh3);
  const long rowStride4 = TOTAL_COLS / 4;

  float acc[8];
#pragma unroll
  for (int e = 0; e < 8; ++e) acc[e] = 0.f;

  for (int i = 0; i < 32; ++i) {
    const long base4 = (offW + (long)(i0 + i) * Hout + j0) >> 2;
    v8f P = {0.f, 0.f, 0.f, 0.f, 0.f, 0.f, 0.f, 0.f};

#if USE_ASYNC_LDS
    asyncB(W4, base4, rowStride4, Bs, t);
    for (int kp = 0; kp < 4; ++kp) {
      float* Bp = Bs + (kp & 1) * 128 * B_STR;
      if (kp < 3) {
        asyncB(W4, base4 + (long)(kp + 1) * 128 * rowStride4, rowStride4,
               Bs + ((kp + 1) & 1) * 128 * B_STR, t);
        WAIT_ASYNC(4);
      } else {
        WAIT_ASYNC(0);
      }
      __syncthreads();
      P = kpanel128(As + kp * 128, Bp, rowA, hi, nC, P);
      __syncthreads();
    }
#else
    PanelRegs pre = fetchB(W4, base4, rowStride4, t);
    for (int kp = 0; kp < 4; ++kp) {
      float* Bp = Bs + (kp & 1) * 128 * B_STR;
      __syncthreads();
      storeB(Bp, t, pre);
      __syncthreads();
      if (kp < 3)
        pre = fetchB(W4, base4 + (long)(kp + 1) * 128 * rowStride4, rowStride4, t);
      P = kpanel128(As + kp * 128, Bp, rowA, hi, nC, P);
    }
#endif

    const float bsc = bts[i * 32 + nC];
#pragma unroll
    for (int e = 0; e < 8; ++e) {
      const float hv = Hs[(rs * 16 + e + 8 * hi) * H_STR + i];
      acc[e] += hv * (P[e] + bsc);
    }
  }

  const int cz = blockIdx.z;
#pragma unroll
  for (int e = 0; e < 8; ++e) {
    const int row = b0 + rs * 16 + e + 8 * hi;
    part[((long)cz * 512 + row) * Hout + (j0 + nC)] = acc[e];
  }
}

// ---------------------------------------------------------------------------
// h_next = relu(sum_c part + bvec)   (or sigmoid -> output for last layer)
// ---------------------------------------------------------------------------
__global__ __launch_bounds__(256) void k_reduce(const float* __restrict__ part,
                                                const float* __restrict__ bvec,
                                                int Hout,
                                                float* __restrict__ hout,
                                                float* __restrict__ sig)
{
  const int b = blockIdx.x, t = threadIdx.x;
  for (int j = t; j < Hout; j += 256) {
    float s = bvec[(long)b * Hout + j];
#pragma unroll
    for (int c = 0; c < 8; ++c)
      s += part[((long)c * 512 + b) * Hout + j];
    if (sig) sig[b * Hout + j]  = 1.f / (1.f + __expf(-s));
    else     hout[b * Hout + j] = fmaxf(s, 0.f);
  }
}

// ---------------------------------------------------------------------------
extern "C" void kernel_launch(void* const* d_in, const int* in_sizes, int n_in,
                              void* d_out, int out_size, void* d_ws, size_t ws_size,
                              hipStream_t stream)
{
  const float* x   = (const float*)d_in[0];
  const float* Wf  = (const float*)d_in[1];
  const float* bf  = (const float*)d_in[2];
  const float* Wh1 = (const float*)d_in[3];
  const float* bh1 = (const float*)d_in[4];
  const float* Wh2 = (const float*)d_in[5];
  const float* bh2 = (const float*)d_in[6];
  const float* Wh3 = (const float*)d_in[7];
  const float* bh3 = (const float*)d_in[8];
  float* out = (float*)d_out;

  float* ws   = (float*)d_ws;
  float* h2   = ws;                    // 512*512
  float* hA   = h2 + 512 * 512;        // 512*256
  float* hB   = hA + 512 * 256;        // 512*256
  float* bvec = hB + 512 * 256;        // 512*256
  float* part = bvec + 512 * 256;      // 8*512*256

  const size_t smL  = (size_t)(64 * A_STR + 2 * 128 * B_STR + 64 * H_STR + 32 * 32) * 4;
  const size_t smH2 = (size_t)(64 * A_STR + 2 * 128 * B_STR + 64) * 4;

  k_feat<<<dim3(512), dim3(256), 0, stream>>>(x, Wf, bf, hA);
  k_h2  <<<dim3(8, 16), dim3(256), smH2, stream>>>(x, Wh1, bh1, Wh2, bh2, h2);

  float* hin = hA;
  float* hout = hB;
  for (int l = 0; l < 3; ++l) {
    const long offW = (long)l * 65792L;
    const long offb = offW + 65536L;
    k_bvec <<<dim3(512), dim3(256), 0, stream>>>(h2, Wh3, bh3, offb, 256, bvec);
    k_layer<<<dim3(8, 8, 8), dim3(256), smL, stream>>>(h2, hin, Wh3, bh3, part,
                                                       256, 256, offW);
    k_reduce<<<dim3(512), dim3(256), 0, stream>>>(part, bvec, 256, hout, nullptr);
    float* tmp = hin; hin = hout; hout = tmp;
  }
  const long offW = 3L * 65792L;          // 197376
  const long offb = offW + 256L * 128L;   // 230144
  k_bvec <<<dim3(512), dim3(256), 0, stream>>>(h2, Wh3, bh3, offb, 128, bvec);
  k_layer<<<dim3(8, 4, 8), dim3(256), smL, stream>>>(h2, hin, Wh3, bh3, part,
                                                     256, 128, offW);
  k_reduce<<<dim3(512), dim3(256), 0, stream>>>(part, bvec, 128, nullptr, out);
}